// KSoftDTW_32152125178375
// MI455X (gfx1250) — compile-verified
//
#include <hip/hip_runtime.h>
#include <hip/hip_bf16.h>
#include <stdint.h>

// Soft-DTW forward, B=64 batches of 512x512, gamma = 1, INF = 1e8.
// One workgroup (512 threads = 16 wave32) per batch; anti-diagonal wavefront.
// CDNA5 paths used:
//   - global_load_async_to_lds_b32 (ASYNCcnt) prefetch ring for the
//     anti-diagonal D reads (hopelessly uncoalesced as direct loads)
//   - s_wait_asynccnt / s_wait_dscnt split-counter waits
//   - hand-split s_barrier_signal/-wait with useful work overlapped between
//     them (async issue + b128 nontemporal output store)

#define N_ROWS 512
#define N_COLS 512
#define RING   16     // ring slots per thread (power of two)
#define RSTR   17     // ring row stride in floats: RSTR odd => bank-
                      // conflict-free skewed per-lane read (17*i + c) % 64
#define LEAD   8      // async prefetch distance in diagonal steps
#define INF_   1e8f

typedef float f4 __attribute__((ext_vector_type(4)));

struct SDTWShared {
  float ring[N_ROWS * RSTR];   // 34816 B : per-row async D staging ring
  float line[3][N_ROWS];       //  6144 B : R values of last 3 diagonals
};

__device__ __forceinline__ unsigned lds_off(const void* p) {
  // gfx1250: generic pointers into LDS carry the LDS byte offset in the low
  // 32 bits (aperture hi-bits | AS3 offset) — what async-to-LDS VDST wants.
  return (unsigned)(uintptr_t)p;
}

__device__ __forceinline__ void async_load_d_b32(unsigned lds_byte_addr,
                                                 unsigned global_byte_off,
                                                 const float* sbase) {
  // GLOBAL_LOAD_ASYNC_TO_LDS_B32, GVS mode: saddr(64b SGPR pair) + voffset.
  asm volatile("global_load_async_to_lds_b32 %0, %1, %2"
               :: "v"(lds_byte_addr), "v"(global_byte_off), "s"(sbase)
               : "memory");
}

__global__ __launch_bounds__(N_ROWS, 1)
void sdtw_fwd_kernel(const float* __restrict__ D, float* __restrict__ R)
{
  __shared__ SDTWShared sm;
  const int b = blockIdx.x;
  const int i = threadIdx.x;                       // my row (0..511)
  const float* Db = D + (size_t)b * (size_t)(N_ROWS * N_COLS);
  float*       Rb = R + (size_t)b * (size_t)(N_ROWS * N_COLS);

  const unsigned ringRow = lds_off(&sm.ring[0]) + (unsigned)(i * RSTR) * 4u;

  // ---- prime the async pipeline: slots for diagonal steps 0..LEAD-1 ----
#pragma unroll
  for (int t = 0; t < LEAD; ++t) {
    int jf = t - i;                                // column needed at step t
    jf = jf < 0 ? 0 : (jf > N_COLS - 1 ? N_COLS - 1 : jf);   // clamp in-bounds
    const unsigned voff = (unsigned)(i * N_COLS + jf) * 4u;
    const unsigned ldst = ringRow + (unsigned)(t & (RING - 1)) * 4u;
    async_load_d_b32(ldst, voff, Db);
  }

  float rL = INF_;                 // R[i][j-1] (my previous output)
  float o0 = 0.f, o1 = 0.f, o2 = 0.f, o3 = 0.f;   // output store coalescer
  int w0 = 0, w1 = 2, w2 = 1;      // line bufs: write t%3, read (t-1)%3, (t-2)%3

  for (int t = 0; t < N_ROWS + N_COLS - 1; ++t) {
    // Slot for step t was issued at iteration t-LEAD (or in the prime loop);
    // exactly 7 younger async issues exist at this point, and completions are
    // in-order per wave, so <=7 outstanding proves our slot has landed.
    asm volatile("s_wait_asynccnt 7" ::: "memory");

    const int  j      = t - i;
    const bool active = ((unsigned)j < (unsigned)N_COLS);

    // ---- gather the three softmin inputs ----
    float above, diag;
    if (i == 0) {
      above = INF_;                                 // Rp[0][j+1]
      diag  = (j == 0) ? 0.0f : INF_;               // Rp[0][j], Rp[0][0] = 0
    } else {
      above = sm.line[w1][i - 1];                   // R[i-1][j]   (step t-1)
      diag  = (j == 0) ? INF_ : sm.line[w2][i - 1]; // R[i-1][j-1] (step t-2)
    }
    const float left = (j == 0) ? INF_ : rL;        // R[i][j-1]

    const float Dv = sm.ring[i * RSTR + (t & (RING - 1))];

    // ---- softmin with gamma = 1 (max-shifted) ----
    const float z0 = -diag, z1 = -above, z2 = -left;
    const float zm = fmaxf(z0, fmaxf(z1, z2));
    const float ssum = __expf(z0 - zm) + __expf(z1 - zm) + __expf(z2 - zm);
    const float val = Dv - (__logf(ssum) + zm);

    // Publish my cell to the diagonal line, then split-barrier: signal as
    // soon as the DS write is complete, overlap independent work, then wait.
    sm.line[w0][i] = val;           // inactive lanes' values never consumed
    asm volatile("s_wait_dscnt 0x0" ::: "memory");
    asm volatile("s_barrier_signal -1" ::: "memory");

    // ======== overlapped with the other 15 waves reaching the barrier ======
    {   // issue async prefetch of D for step t+LEAD (uniform EXEC)
      int jf = t + LEAD - i;
      jf = jf < 0 ? 0 : (jf > N_COLS - 1 ? N_COLS - 1 : jf);
      const unsigned voff = (unsigned)(i * N_COLS + jf) * 4u;
      const unsigned ldst = ringRow + (unsigned)((t + LEAD) & (RING - 1)) * 4u;
      async_load_d_b32(ldst, voff, Db);
    }
    if (active) {
      rL = val;
      o0 = o1; o1 = o2; o2 = o3; o3 = val;          // register shift chain
      if ((j & 3) == 3) {                           // 16B-aligned b128 store
        f4 v = {o0, o1, o2, o3};
        __builtin_nontemporal_store(v, (f4*)(Rb + i * N_COLS + (j - 3)));
      }
    }
    { const int tmp = w2; w2 = w1; w1 = w0; w0 = tmp; }  // rotate line bufs
    // =======================================================================

    asm volatile("s_barrier_wait -1" ::: "memory");
  }
}

extern "C" void kernel_launch(void* const* d_in, const int* in_sizes, int n_in,
                              void* d_out, int out_size, void* d_ws, size_t ws_size,
                              hipStream_t stream) {
  (void)n_in; (void)out_size; (void)d_ws; (void)ws_size;
  const float* D = (const float*)d_in[0];
  float* out = (float*)d_out;
  const int B = in_sizes[0] / (N_ROWS * N_COLS);   // 64
  sdtw_fwd_kernel<<<dim3(B), dim3(N_ROWS), 0, stream>>>(D, out);
}